// AdaHister_7842610283315
// MI455X (gfx1250) — compile-verified
//
#include <hip/hip_runtime.h>

#define LEVEL 255

typedef __attribute__((ext_vector_type(2))) float v2f;
typedef __attribute__((ext_vector_type(8))) float v8f;

// bin = clamp(ceil(255*v) - 1, 0, 254), matching searchsorted(step, v, 'left')-1
// against step = linspace(0,1,256): smallest j with j/255 >= v, minus 1.
__device__ __forceinline__ int bin_of(float v) {
    v = fminf(fmaxf(v, 0.0f), 1.0f);
    int j = (int)ceilf(v * 255.0f);
    return min(max(j - 1, 0), LEVEL - 1);
}

// ---------------------------------------------------------------------------
// Kernel 0: zero the per-batch histogram counters in workspace.
// ---------------------------------------------------------------------------
__global__ void zero_counts_kernel(unsigned int* counts, int n) {
    int i = blockIdx.x * blockDim.x + threadIdx.x;
    if (i < n) counts[i] = 0u;
}

// ---------------------------------------------------------------------------
// Kernel 1: per-batch histogram. Per-wave privatized LDS histograms (8 waves
// x 256 bins = 8KB of the 320KB WGP LDS) -> ds_add_u32 atomics, then one
// global_atomic_add_u32 per bin per block.
// ---------------------------------------------------------------------------
__global__ __launch_bounds__(256) void hist_kernel(const float4* __restrict__ in,
                                                   unsigned int* __restrict__ counts,
                                                   int vec_per_batch,
                                                   int blocks_per_batch) {
    __shared__ unsigned int h[8][256];
    const int b    = blockIdx.y;
    const int wave = threadIdx.x >> 5;   // wave32: 8 waves per 256-thread block

    for (int i = threadIdx.x; i < 8 * 256; i += blockDim.x)
        ((unsigned int*)h)[i] = 0u;
    __syncthreads();

    const float4* base = in + (size_t)b * vec_per_batch;
    const int stride = blocks_per_batch * blockDim.x;
    for (int i = blockIdx.x * blockDim.x + threadIdx.x; i < vec_per_batch; i += stride) {
        float4 v = base[i];
        atomicAdd(&h[wave][bin_of(v.x)], 1u);
        atomicAdd(&h[wave][bin_of(v.y)], 1u);
        atomicAdd(&h[wave][bin_of(v.z)], 1u);
        atomicAdd(&h[wave][bin_of(v.w)], 1u);
    }
    __syncthreads();

    // threadIdx.x indexes the bin (blockDim.x == 256)
    unsigned int s = 0;
    #pragma unroll
    for (int w = 0; w < 8; ++w) s += h[w][threadIdx.x];
    if (s) atomicAdd(&counts[b * 256 + threadIdx.x], s);
}

// ---------------------------------------------------------------------------
// Kernel 2: pdf -> cdf via V_WMMA_F32_16X16X4_F32 (full f32 precision, RNE).
// One wave32 per batch (EXEC all-ones through the WMMA region).
//
//   P[r][c] = pdf[16r + c]  (16x16)
//   R = P x U  with U upper-triangular ones   -> within-row inclusive prefix
//   K=16 split into 4 chained K=4 WMMAs accumulating into the f32 C/D regs.
//   cdf[16r+c] = R[r][c] + sum_{s<r} R[s][15]  (row offsets via tiny LDS pass)
//
// Layout assumptions (ISA 7.12.2, 32-bit A 16x4 / B 4x16 mirrored):
//   A: lane m=lane&15 holds row M=m; kh=lane>>4; v0 -> K=4q+2kh, v1 -> K=4q+2kh+1
//   B: lane n=lane&15 holds col N=n; v0 -> row K=4q+2kh, v1 -> row K=4q+2kh+1
//   C/D: vgpr v, lanes 0-15 -> M=v, lanes 16-31 -> M=v+8, N=lane&15
// ---------------------------------------------------------------------------
__global__ __launch_bounds__(32) void cdf_kernel(const unsigned int* __restrict__ counts,
                                                 float* __restrict__ cdf,
                                                 float invN) {
    __shared__ float p[256];
    __shared__ float r[256];
    __shared__ float o_s[16];

    const int b    = blockIdx.x;
    const int lane = threadIdx.x;          // 0..31
    const int m    = lane & 15;            // A row / B col / D col
    const int kh   = lane >> 4;            // K half-select

    // pdf, padded with 0 at bin 255 (uniform trip count -> no divergence)
    #pragma unroll
    for (int k = lane; k < 256; k += 32) {
        float v = (k < LEVEL) ? ((float)counts[b * 256 + k]) * invN : 0.0f;
        p[k] = v;
    }
    __syncthreads();

    v8f acc = {};
    #pragma unroll
    for (int q = 0; q < 4; ++q) {
        const int k0 = 4 * q + 2 * kh;
        v2f a, bm;
        a.x  = p[16 * m + k0];
        a.y  = p[16 * m + k0 + 1];
        bm.x = (k0 <= m) ? 1.0f : 0.0f;       // U[k0][n],  n == m == lane&15
        bm.y = (k0 + 1 <= m) ? 1.0f : 0.0f;   // U[k0+1][n]
        // (neg_a, A, neg_b, B, c_mod, C, reuse_a, reuse_b)
        acc = __builtin_amdgcn_wmma_f32_16x16x4_f32(
            false, a, false, bm, (short)0, acc, false, false);
    }

    // scatter D (16x16 row-prefix matrix R) into LDS per documented C/D layout
    #pragma unroll
    for (int v = 0; v < 8; ++v) {
        const int M = (lane < 16) ? v : v + 8;
        r[16 * M + m] = acc[v];
    }
    __syncthreads();

    // exclusive prefix of row totals t[rr] = R[rr][15] (tiny, LDS-broadcast)
    if (lane < 16) {
        float run = 0.0f;
        for (int rr = 0; rr < lane; ++rr) run += r[16 * rr + 15];
        o_s[lane] = run;
    }
    __syncthreads();

    #pragma unroll
    for (int k = lane; k < 256; k += 32)
        cdf[b * 256 + k] = r[k] + o_s[k >> 4];
}

// ---------------------------------------------------------------------------
// Kernel 3: apply. cdf table (1KB) broadcast to LDS; float4 streaming pass.
// Second read of the input hits the 192MB L2 (input is 100MB, L2-resident).
// ---------------------------------------------------------------------------
__global__ __launch_bounds__(256) void apply_kernel(const float4* __restrict__ in,
                                                    const float* __restrict__ cdf,
                                                    float4* __restrict__ out,
                                                    int vec_per_batch,
                                                    int blocks_per_batch) {
    __shared__ float c[256];
    const int b = blockIdx.y;
    c[threadIdx.x] = cdf[b * 256 + threadIdx.x];   // blockDim.x == 256
    __syncthreads();

    const float4* ibase = in  + (size_t)b * vec_per_batch;
    float4*       obase = out + (size_t)b * vec_per_batch;
    const int stride = blocks_per_batch * blockDim.x;
    for (int i = blockIdx.x * blockDim.x + threadIdx.x; i < vec_per_batch; i += stride) {
        float4 v = ibase[i];
        float4 o;
        o.x = c[bin_of(v.x)];
        o.y = c[bin_of(v.y)];
        o.z = c[bin_of(v.z)];
        o.w = c[bin_of(v.w)];
        obase[i] = o;
    }
}

extern "C" void kernel_launch(void* const* d_in, const int* in_sizes, int n_in,
                              void* d_out, int out_size, void* d_ws, size_t ws_size,
                              hipStream_t stream) {
    const float* in  = (const float*)d_in[0];
    float*       out = (float*)d_out;

    const int total = in_sizes[0];          // 32*3*512*512 = 25,165,824
    const int B = 32;
    const int N = total / B;                // 786,432 elements per batch
    const int vec_per_batch = N / 4;        // 196,608 float4 per batch

    // workspace layout: [counts: B*256 u32][cdf: B*256 f32] = 64KB
    unsigned int* counts = (unsigned int*)d_ws;
    float*        cdf    = (float*)((char*)d_ws + (size_t)B * 256 * sizeof(unsigned int));

    zero_counts_kernel<<<(B * 256 + 255) / 256, 256, 0, stream>>>(counts, B * 256);

    const int bpbH = 96;   // 96*32 = 3072 workgroups; 8 float4 per thread
    hist_kernel<<<dim3(bpbH, B), 256, 0, stream>>>((const float4*)in, counts,
                                                   vec_per_batch, bpbH);

    cdf_kernel<<<B, 32, 0, stream>>>(counts, cdf, 1.0f / (float)N);

    const int bpbA = 128;  // 128*32 = 4096 workgroups; 6 float4 per thread
    apply_kernel<<<dim3(bpbA, B), 256, 0, stream>>>((const float4*)in, cdf,
                                                    (float4*)out, vec_per_batch, bpbA);
}